// GNN_Autoencoder_89970974916698
// MI455X (gfx1250) — compile-verified
//
#include <hip/hip_runtime.h>

typedef __attribute__((ext_vector_type(2))) float v2f;
typedef __attribute__((ext_vector_type(8))) float v8f;

#define N_NODES 100000
#define N_EDGES 1600000
#define N_ETOT  (N_EDGES + N_NODES)   // edges + self loops
#define F_IN    64
#define F_HID   16

// ---------------- degree / norm prep ----------------

__global__ void k_init_deg(float* __restrict__ deg) {
    int v = blockIdx.x * blockDim.x + threadIdx.x;
    if (v < N_NODES) deg[v] = 1.0f;   // self-loop contribution
}

__global__ void k_deg_accum(const int* __restrict__ eidx, float* __restrict__ deg) {
    int e = blockIdx.x * blockDim.x + threadIdx.x;
    if (e < N_EDGES) {
        int v = eidx[N_EDGES + e];    // dst row of edge_index
        atomicAdd(&deg[v], 1.0f);
    }
}

__global__ void k_dinv(float* __restrict__ deg) {
    int v = blockIdx.x * blockDim.x + threadIdx.x;
    if (v < N_NODES) deg[v] = __frsqrt_rn(deg[v]);  // in place: deg -> deg^-1/2
}

__global__ void k_norm(const int* __restrict__ eidx, const float* __restrict__ dinv,
                       float* __restrict__ norm) {
    int e = blockIdx.x * blockDim.x + threadIdx.x;
    if (e >= N_ETOT) return;
    if (e < N_EDGES) {
        int u = eidx[e];
        int v = eidx[N_EDGES + e];
        norm[e] = dinv[u] * dinv[v];
    } else {
        float d = dinv[e - N_EDGES];
        norm[e] = d * d;              // self loop: dinv[v]^2
    }
}

// ---------------- dense transforms via V_WMMA_F32_16X16X4_F32 ----------------
// One wave -> one 16x16 output tile. 100000 rows = 6250 exact row tiles.

__global__ void k_gemm1(const float* __restrict__ x, const float* __restrict__ W1,
                        float* __restrict__ h) {
    unsigned gtid = blockIdx.x * blockDim.x + threadIdx.x;
    unsigned wave = gtid >> 5;
    unsigned lane = threadIdx.x & 31u;
    if (wave >= (N_NODES / 16)) return;       // uniform per wave: EXEC stays all-1
    const int row0 = (int)wave * 16;
    const int m  = lane & 15;                 // M for A, N for B/C
    const int hi = lane >> 4;                 // lane half select
    const float* xrow = x + (long)(row0 + m) * F_IN;
    v8f acc = {};
#pragma unroll
    for (int kk = 0; kk < F_IN; kk += 4) {
        v2f a, b;
        // A 16x4: VGPR r holds K = kk + 2*hi + r  (consecutive -> float2-able)
        a.x = xrow[kk + 2 * hi + 0];
        a.y = xrow[kk + 2 * hi + 1];
        // B 4x16: VGPR r holds row K = kk + 2*hi + r, col N = m
        b.x = W1[(kk + 2 * hi + 0) * F_HID + m];
        b.y = W1[(kk + 2 * hi + 1) * F_HID + m];
        acc = __builtin_amdgcn_wmma_f32_16x16x4_f32(
                  false, a, false, b, (short)0, acc, false, false);
    }
    // C/D layout: VGPR r -> row (row0 + r + 8*hi), col m
#pragma unroll
    for (int r = 0; r < 8; ++r)
        h[(long)(row0 + r + 8 * hi) * F_HID + m] = acc[r];
}

__global__ void k_gemm2(const float* __restrict__ z, const float* __restrict__ W2,
                        float* __restrict__ h2) {
    unsigned gtid = blockIdx.x * blockDim.x + threadIdx.x;
    unsigned wave = gtid >> 5;
    unsigned lane = threadIdx.x & 31u;
    if (wave >= (N_NODES / 16) * 4) return;   // 4 column tiles of 16 in F_IN=64
    const int nt   = (int)(wave & 3u);
    const int row0 = (int)(wave >> 2) * 16;
    const int n0   = nt * 16;
    const int m  = lane & 15;
    const int hi = lane >> 4;
    const float* zrow = z + (long)(row0 + m) * F_HID;
    v8f acc = {};
#pragma unroll
    for (int kk = 0; kk < F_HID; kk += 4) {
        v2f a, b;
        a.x = zrow[kk + 2 * hi + 0];
        a.y = zrow[kk + 2 * hi + 1];
        b.x = W2[(kk + 2 * hi + 0) * F_IN + n0 + m];
        b.y = W2[(kk + 2 * hi + 1) * F_IN + n0 + m];
        acc = __builtin_amdgcn_wmma_f32_16x16x4_f32(
                  false, a, false, b, (short)0, acc, false, false);
    }
#pragma unroll
    for (int r = 0; r < 8; ++r)
        h2[(long)(row0 + r + 8 * hi) * F_IN + n0 + m] = acc[r];
}

// ---------------- bias init + edge scatter ----------------

__global__ void k_init_bias(float* __restrict__ out, const float* __restrict__ b,
                            int fmask, int total) {
    int tid = blockIdx.x * blockDim.x + threadIdx.x;
    if (tid < total) out[tid] = b[tid & fmask];
}

// one thread per (edge, feature); F = 1<<logF features per row
__global__ void k_scatter(const int* __restrict__ eidx, const float* __restrict__ norm,
                          const float* __restrict__ hsrc, float* __restrict__ out,
                          int logF, int total) {
    int tid = blockIdx.x * blockDim.x + threadIdx.x;
    if (tid >= total) return;
    int e = tid >> logF;
    int f = tid & ((1 << logF) - 1);
    int u, v;
    if (e < N_EDGES) { u = eidx[e]; v = eidx[N_EDGES + e]; }
    else             { u = v = e - N_EDGES; }
    float msg = hsrc[((long)u << logF) + f] * norm[e];
    atomicAdd(&out[((long)v << logF) + f], msg);
}

// ---------------- launcher ----------------

extern "C" void kernel_launch(void* const* d_in, const int* in_sizes, int n_in,
                              void* d_out, int out_size, void* d_ws, size_t ws_size,
                              hipStream_t stream) {
    const float* x    = (const float*)d_in[0];
    const int*   eidx = (const int*)d_in[1];   // edge_index [2, N_EDGES]
    const float* W1   = (const float*)d_in[2];
    const float* b1   = (const float*)d_in[3];
    const float* W2   = (const float*)d_in[4];
    const float* b2   = (const float*)d_in[5];
    float* recon = (float*)d_out;

    // workspace layout (floats): dinv[n] | norm[E+n] | z[16n] | h / h2 [64n]
    float* dinv = (float*)d_ws;
    float* norm = dinv + N_NODES;
    float* z    = norm + N_ETOT;
    float* h    = z + (long)N_NODES * F_HID;   // reused: h (16n) then h2 (64n)

    const int B = 256;
    // 1) degree (with self loops) -> dinv
    k_init_deg <<<(N_NODES + B - 1) / B, B, 0, stream>>>(dinv);
    k_deg_accum<<<(N_EDGES + B - 1) / B, B, 0, stream>>>(eidx, dinv);
    k_dinv     <<<(N_NODES + B - 1) / B, B, 0, stream>>>(dinv);
    // 2) per-edge norms (shared by both layers)
    k_norm     <<<(N_ETOT + B - 1) / B, B, 0, stream>>>(eidx, dinv, norm);

    // 3) encoder: h = x @ W1 (WMMA), z = b1 + scatter(norm * h[src])
    {
        int waves = N_NODES / 16;                       // 6250
        k_gemm1<<<(waves * 32 + B - 1) / B, B, 0, stream>>>(x, W1, h);
        int totz = N_NODES * F_HID;
        k_init_bias<<<(totz + B - 1) / B, B, 0, stream>>>(z, b1, F_HID - 1, totz);
        int tot = N_ETOT * F_HID;                       // 27.2M
        k_scatter<<<(tot + B - 1) / B, B, 0, stream>>>(eidx, norm, h, z, 4, tot);
    }

    // 4) decoder: h2 = z @ W2 (WMMA), recon = b2 + scatter(norm * h2[src])
    {
        int waves = (N_NODES / 16) * 4;                 // 25000
        k_gemm2<<<(waves * 32 + B - 1) / B, B, 0, stream>>>(z, W2, h);
        int totr = N_NODES * F_IN;
        k_init_bias<<<(totr + B - 1) / B, B, 0, stream>>>(recon, b2, F_IN - 1, totr);
        int tot = N_ETOT * F_IN;                        // 108.8M
        k_scatter<<<(tot + B - 1) / B, B, 0, stream>>>(eidx, norm, h, recon, 6, tot);
    }
}